// TransformerConv_75522704933007
// MI455X (gfx1250) — compile-verified
//
#include <hip/hip_runtime.h>

typedef __attribute__((ext_vector_type(2))) float v2f;
typedef __attribute__((ext_vector_type(8))) float v8f;

#define KNN_K 16

// ---------------------------------------------------------------------------
// Brute-force kNN (k=16) within each batch, LDS-tiled candidates, sorted
// top-16 kept in registers (fully unrolled insertion).
// ---------------------------------------------------------------------------
__global__ void knn_kernel(const float* __restrict__ pts, int* __restrict__ nbr, int Nper) {
  const int TILE = 256;
  __shared__ float sx[TILE], sy[TILE], sz[TILE];
  int blocksPerBatch = Nper / TILE;
  int batch = blockIdx.x / blocksPerBatch;
  int row   = (blockIdx.x % blocksPerBatch) * TILE + threadIdx.x; // in-batch row
  int base  = batch * Nper;
  float px = pts[(base + row) * 3 + 0];
  float py = pts[(base + row) * 3 + 1];
  float pz = pts[(base + row) * 3 + 2];
  float dist[KNN_K];
  int   idx[KNN_K];
#pragma unroll
  for (int s = 0; s < KNN_K; ++s) { dist[s] = 3.0e38f; idx[s] = 0; }

  for (int c0 = 0; c0 < Nper; c0 += TILE) {
    __syncthreads();
    int c = c0 + threadIdx.x;
    sx[threadIdx.x] = pts[(base + c) * 3 + 0];
    sy[threadIdx.x] = pts[(base + c) * 3 + 1];
    sz[threadIdx.x] = pts[(base + c) * 3 + 2];
    __syncthreads();
    for (int j = 0; j < TILE; ++j) {
      float dx = px - sx[j], dy = py - sy[j], dz = pz - sz[j];
      float d2 = dx * dx + dy * dy + dz * dz;
      int cand = c0 + j;
      if (cand != row && d2 < dist[KNN_K - 1]) {
        float dv = d2; int iv = cand;
#pragma unroll
        for (int s = 0; s < KNN_K; ++s) {
          if (dv < dist[s]) {
            float td = dist[s]; int ti = idx[s];
            dist[s] = dv; idx[s] = iv;
            dv = td; iv = ti;
          }
        }
      }
    }
  }
#pragma unroll
  for (int s = 0; s < KNN_K; ++s)
    nbr[(base + row) * KNN_K + s] = base + idx[s];  // global node ids
}

// ---------------------------------------------------------------------------
// Aligned GEMM: Y[M,N] = X[M,K] @ W[K,N] (+bias), with M%16==0, K%4==0,
// N == NT*16 (compile-time). One wave computes a 16 x (NT*16) strip: the A
// fragment is loaded once per K-step and reused by NT WMMAs. No guards in
// the K-loop -> no exec-mask traffic, loads pair into b64.
// ---------------------------------------------------------------------------
template <int NT>
__global__ void gemm_wmma_aligned(const float* __restrict__ X, const float* __restrict__ W,
                                  const float* __restrict__ bias, float* __restrict__ Y,
                                  int M, int K, int hasBias) {
  const int N = NT * 16;
  int wpb  = blockDim.x >> 5;
  int tile = blockIdx.x * wpb + (threadIdx.x >> 5);   // 16-row strip index
  if (tile * 16 >= M) return;                         // wave-uniform
  int lane = threadIdx.x & 31;
  int half = lane >> 4, l = lane & 15;
  int row = tile * 16 + l;

  v8f acc[NT];
#pragma unroll
  for (int t = 0; t < NT; ++t) acc[t] = (v8f){};

  for (int k0 = 0; k0 < K; k0 += 4) {
    int ka = k0 + 2 * half;
    v2f a;
    a.x = X[row * K + ka];
    a.y = X[row * K + ka + 1];
#pragma unroll
    for (int t = 0; t < NT; ++t) {
      int col = t * 16 + l;
      v2f b;
      b.x = W[ka * N + col];
      b.y = W[(ka + 1) * N + col];
      acc[t] = __builtin_amdgcn_wmma_f32_16x16x4_f32(false, a, false, b,
                                                     (short)0, acc[t], false, false);
    }
  }
#pragma unroll
  for (int t = 0; t < NT; ++t) {
    int col = t * 16 + l;
    float bb = hasBias ? bias[col] : 0.0f;
#pragma unroll
    for (int r = 0; r < 8; ++r)
      Y[(tile * 16 + half * 8 + r) * N + col] = acc[t][r] + bb;
  }
}

// ---------------------------------------------------------------------------
// Ragged GEMM (K=67/35/3, N=3): branchless zero-padding. Loads are issued
// unconditionally at clamped (always-valid) addresses, then zeroed with
// selects -> no exec-mask save/restore in the K-loop, EXEC stays all-ones
// for the WMMA.
// ---------------------------------------------------------------------------
__global__ void gemm_wmma_ragged(const float* __restrict__ X, const float* __restrict__ W,
                                 const float* __restrict__ bias, float* __restrict__ Y,
                                 int M, int K, int N, int hasBias) {
  int wpb  = blockDim.x >> 5;
  int tile = blockIdx.x * wpb + (threadIdx.x >> 5);
  int mt = (M + 15) >> 4, nt = (N + 15) >> 4;
  if (tile >= mt * nt) return;                        // wave-uniform
  int lane = threadIdx.x & 31;
  int tm = tile / nt, tn = tile % nt;
  int half = lane >> 4, l = lane & 15;
  int row = tm * 16 + l;
  int col = tn * 16 + l;
  bool rowOK = row < M, colOK = col < N;
  int rowc = rowOK ? row : (M - 1);
  int colc = colOK ? col : (N - 1);

  v8f acc = {};
  for (int k0 = 0; k0 < K; k0 += 4) {
    int ka = k0 + 2 * half;
    bool k0ok = ka < K, k1ok = (ka + 1) < K;
    int ka0 = k0ok ? ka : (K - 1);
    int ka1 = k1ok ? (ka + 1) : (K - 1);
    float ax = X[rowc * K + ka0];
    float ay = X[rowc * K + ka1];
    float bx = W[ka0 * N + colc];
    float by = W[ka1 * N + colc];
    v2f a, b;
    a.x = (rowOK && k0ok) ? ax : 0.0f;
    a.y = (rowOK && k1ok) ? ay : 0.0f;
    b.x = (colOK && k0ok) ? bx : 0.0f;
    b.y = (colOK && k1ok) ? by : 0.0f;
    acc = __builtin_amdgcn_wmma_f32_16x16x4_f32(false, a, false, b,
                                                (short)0, acc, false, false);
  }

  if (colOK) {
    float bb = hasBias ? bias[col] : 0.0f;
#pragma unroll
    for (int r = 0; r < 8; ++r) {
      int m = tm * 16 + half * 8 + r;
      if (m < M) Y[m * N + col] = acc[r] + bb;
    }
  }
}

// ---------------------------------------------------------------------------
// Per-node attention scores + softmax over k=16 neighbors.
// ---------------------------------------------------------------------------
template <int D>
__global__ void attn_scores_kernel(const float* __restrict__ Q, const float* __restrict__ Kf,
                                   const int* __restrict__ nbr, float* __restrict__ alpha,
                                   int Nt) {
  int n = blockIdx.x * blockDim.x + threadIdx.x;
  if (n >= Nt) return;
  float q[D];
#pragma unroll
  for (int t = 0; t < D; ++t) q[t] = Q[n * D + t];
  const float scale = rsqrtf((float)D);
  float sc[KNN_K];
  float mx = -3.0e38f;
#pragma unroll
  for (int j = 0; j < KNN_K; ++j) {
    int m = nbr[n * KNN_K + j];
    float s = 0.0f;
#pragma unroll
    for (int t = 0; t < D; ++t) s += q[t] * Kf[m * D + t];
    s *= scale;
    sc[j] = s;
    mx = fmaxf(mx, s);
  }
  float sum = 0.0f;
#pragma unroll
  for (int j = 0; j < KNN_K; ++j) { sc[j] = __expf(sc[j] - mx); sum += sc[j]; }
  float inv = 1.0f / sum;
#pragma unroll
  for (int j = 0; j < KNN_K; ++j) alpha[n * KNN_K + j] = sc[j] * inv;
}

// ---------------------------------------------------------------------------
// out[n,t] = sum_j alpha[n,j] * V[nbr[n,j], t] + skip[n,t]
// ---------------------------------------------------------------------------
__global__ void attn_apply_kernel(const float* __restrict__ V, const float* __restrict__ skip,
                                  const float* __restrict__ alpha, const int* __restrict__ nbr,
                                  float* __restrict__ out, int Nt, int D) {
  int i = blockIdx.x * blockDim.x + threadIdx.x;
  if (i >= Nt * D) return;
  int n = i / D, t = i - n * D;
  float acc = skip[i];
#pragma unroll
  for (int j = 0; j < KNN_K; ++j) {
    int m = nbr[n * KNN_K + j];
    acc += alpha[n * KNN_K + j] * V[m * D + t];
  }
  out[i] = acc;
}

// ---------------------------------------------------------------------------
// nearest upsample x3 along node axis (row-repeat works batch-globally)
// ---------------------------------------------------------------------------
__global__ void upsample3_kernel(const float* __restrict__ in, float* __restrict__ out,
                                 int Nin, int C) {
  int i = blockIdx.x * blockDim.x + threadIdx.x;
  if (i >= Nin * 3 * C) return;
  int c = i % C;
  int r = i / C;
  out[i] = in[(r / 3) * C + c];
}

// ---------------------------------------------------------------------------
// cat = [p(3) | feat(C)]  -> [Nt x (3+C)]
// ---------------------------------------------------------------------------
__global__ void concat_pb_kernel(const float* __restrict__ p, const float* __restrict__ f,
                                 float* __restrict__ out, int Nt, int C) {
  int Wd = 3 + C;
  int i = blockIdx.x * blockDim.x + threadIdx.x;
  if (i >= Nt * Wd) return;
  int n = i / Wd, c = i - n * Wd;
  out[i] = (c < 3) ? p[n * 3 + c] : f[n * C + (c - 3)];
}

// ---------------------------------------------------------------------------
// host-side orchestration
// ---------------------------------------------------------------------------
static inline int ceildiv(int a, int b) { return (a + b - 1) / b; }

static void launch_gemm(const float* X, const float* W, const float* bias, float* Y,
                        int M, int K, int N, hipStream_t s) {
  const int WPB = 4;
  if ((M % 16) == 0 && (K % 4) == 0 && N == 64) {
    int tiles = M / 16;
    gemm_wmma_aligned<4><<<ceildiv(tiles, WPB), WPB * 32, 0, s>>>(X, W, bias, Y, M, K,
                                                                  bias != nullptr);
  } else if ((M % 16) == 0 && (K % 4) == 0 && N == 32) {
    int tiles = M / 16;
    gemm_wmma_aligned<2><<<ceildiv(tiles, WPB), WPB * 32, 0, s>>>(X, W, bias, Y, M, K,
                                                                  bias != nullptr);
  } else {
    int tiles = ceildiv(M, 16) * ceildiv(N, 16);
    gemm_wmma_ragged<<<ceildiv(tiles, WPB), WPB * 32, 0, s>>>(X, W, bias, Y, M, K, N,
                                                              bias != nullptr);
  }
}

static void launch_scores(const float* Q, const float* Kf, const int* nbr, float* alpha,
                          int Nt, int D, hipStream_t s) {
  int thr = 128, g = ceildiv(Nt, thr);
  if (D == 64)      attn_scores_kernel<64><<<g, thr, 0, s>>>(Q, Kf, nbr, alpha, Nt);
  else if (D == 32) attn_scores_kernel<32><<<g, thr, 0, s>>>(Q, Kf, nbr, alpha, Nt);
  else              attn_scores_kernel<3><<<g, thr, 0, s>>>(Q, Kf, nbr, alpha, Nt);
}

static void tconv(const float* x, const int* nbr, int Nt, int din, int dout,
                  void* const* d_in, int layer,
                  float* q, float* kb, float* v, float* sk, float* alpha, float* out,
                  hipStream_t s) {
  const float* Wq = (const float*)d_in[2 + layer * 5 + 0];
  const float* Wk = (const float*)d_in[2 + layer * 5 + 1];
  const float* Wv = (const float*)d_in[2 + layer * 5 + 2];
  const float* Ws = (const float*)d_in[2 + layer * 5 + 3];
  const float* bb = (const float*)d_in[2 + layer * 5 + 4];
  launch_gemm(x, Wq, nullptr, q,  Nt, din, dout, s);
  launch_gemm(x, Wk, nullptr, kb, Nt, din, dout, s);
  launch_gemm(x, Wv, nullptr, v,  Nt, din, dout, s);
  launch_gemm(x, Ws, bb,      sk, Nt, din, dout, s);  // skip + bias
  launch_scores(q, kb, nbr, alpha, Nt, dout, s);
  int tot = Nt * dout, thr = 256;
  attn_apply_kernel<<<ceildiv(tot, thr), thr, 0, s>>>(v, sk, alpha, nbr, out, Nt, dout);
}

extern "C" void kernel_launch(void* const* d_in, const int* in_sizes, int n_in,
                              void* d_out, int out_size, void* d_ws, size_t ws_size,
                              hipStream_t stream) {
  (void)in_sizes; (void)n_in; (void)out_size; (void)ws_size;
  const int B = 4;
  const int N1 = 1024, Nt1 = B * N1;   // 4096
  const int N2 = 3072, Nt2 = B * N2;   // 12288
  const int N3 = 9216, Nt3 = B * N3;   // 36864

  const float* features = (const float*)d_in[0];
  const float* points   = (const float*)d_in[1];
  float* outp = (float*)d_out;

  // bump allocator over d_ws (256B aligned)
  char* ws = (char*)d_ws;
  size_t off = 0;
  auto alloc = [&](size_t bytes) -> void* {
    void* p = ws + off;
    off += (bytes + 255) & ~(size_t)255;
    return p;
  };
  const size_t QMAX = (size_t)Nt2 * 32;  // 393216 floats: largest q/k/v/skip
  int*   nbr1 = (int*)  alloc((size_t)Nt1 * KNN_K * 4);
  int*   nbr2 = (int*)  alloc((size_t)Nt2 * KNN_K * 4);
  int*   nbr3 = (int*)  alloc((size_t)Nt3 * KNN_K * 4);
  float* qb   = (float*)alloc(QMAX * 4);
  float* kb   = (float*)alloc(QMAX * 4);
  float* vb   = (float*)alloc(QMAX * 4);
  float* sb   = (float*)alloc(QMAX * 4);
  float* alph = (float*)alloc((size_t)Nt3 * KNN_K * 4);
  float* b1a  = (float*)alloc((size_t)Nt1 * 64 * 4);
  float* b1b  = (float*)alloc((size_t)Nt1 * 64 * 4);
  float* cat1 = (float*)alloc((size_t)Nt1 * 67 * 4);
  float* p1   = (float*)alloc((size_t)Nt1 * 3 * 4);
  float* b1u  = (float*)alloc((size_t)Nt2 * 64 * 4);
  float* p1u  = (float*)alloc((size_t)Nt2 * 3 * 4);
  float* b2a  = (float*)alloc((size_t)Nt2 * 32 * 4);
  float* b2b  = (float*)alloc((size_t)Nt2 * 32 * 4);
  float* cat2 = (float*)alloc((size_t)Nt2 * 35 * 4);
  float* p2   = (float*)alloc((size_t)Nt2 * 3 * 4);
  float* b2u  = (float*)alloc((size_t)Nt3 * 32 * 4);
  float* p2u  = (float*)alloc((size_t)Nt3 * 3 * 4);
  float* b3a  = (float*)alloc((size_t)Nt3 * 3 * 4);
  float* b3b  = (float*)alloc((size_t)Nt3 * 3 * 4);

  // ---- stage 1 (Nt=4096, d=64) ----
  knn_kernel<<<B * (N1 / 256), 256, 0, stream>>>(points, nbr1, N1);
  tconv(features, nbr1, Nt1, 64, 64, d_in, 0, qb, kb, vb, sb, alph, b1a, stream);
  tconv(b1a,      nbr1, Nt1, 64, 64, d_in, 1, qb, kb, vb, sb, alph, b1b, stream);
  tconv(b1b,      nbr1, Nt1, 64, 64, d_in, 2, qb, kb, vb, sb, alph, b1a, stream);
  concat_pb_kernel<<<ceildiv(Nt1 * 67, 256), 256, 0, stream>>>(points, b1a, cat1, Nt1, 64);
  tconv(cat1,     nbr1, Nt1, 67, 3,  d_in, 3, qb, kb, vb, sb, alph, p1,  stream);
  upsample3_kernel<<<ceildiv(Nt1 * 3 * 64, 256), 256, 0, stream>>>(b1a, b1u, Nt1, 64);
  upsample3_kernel<<<ceildiv(Nt1 * 3 * 3, 256), 256, 0, stream>>>(p1, p1u, Nt1, 3);

  // ---- stage 2 (Nt=12288, d=32) ----
  knn_kernel<<<B * (N2 / 256), 256, 0, stream>>>(p1u, nbr2, N2);
  tconv(b1u, nbr2, Nt2, 64, 32, d_in, 4, qb, kb, vb, sb, alph, b2a, stream);
  tconv(b2a, nbr2, Nt2, 32, 32, d_in, 5, qb, kb, vb, sb, alph, b2b, stream);
  tconv(b2b, nbr2, Nt2, 32, 32, d_in, 6, qb, kb, vb, sb, alph, b2a, stream);
  concat_pb_kernel<<<ceildiv(Nt2 * 35, 256), 256, 0, stream>>>(p1u, b2a, cat2, Nt2, 32);
  tconv(cat2, nbr2, Nt2, 35, 3, d_in, 7, qb, kb, vb, sb, alph, p2, stream);
  upsample3_kernel<<<ceildiv(Nt2 * 3 * 32, 256), 256, 0, stream>>>(b2a, b2u, Nt2, 32);
  upsample3_kernel<<<ceildiv(Nt2 * 3 * 3, 256), 256, 0, stream>>>(p2, p2u, Nt2, 3);

  // ---- stage 3 (Nt=36864, d=3) ----
  knn_kernel<<<B * (N3 / 256), 256, 0, stream>>>(p2u, nbr3, N3);
  tconv(b2u, nbr3, Nt3, 32, 3, d_in, 8,  qb, kb, vb, sb, alph, b3a, stream);
  tconv(b3a, nbr3, Nt3, 3,  3, d_in, 9,  qb, kb, vb, sb, alph, b3b, stream);
  tconv(b3b, nbr3, Nt3, 3,  3, d_in, 10, qb, kb, vb, sb, alph, outp, stream);
}